// BidirectionalMamba_64347200029127
// MI455X (gfx1250) — compile-verified
//
#include <hip/hip_runtime.h>
#include <hip/hip_bf16.h>
#include <math.h>

// ---------------------------------------------------------------------------
// Bidirectional Mamba on gfx1250 (MI455X), fp32 end-to-end.
//   * GEMMs on V_WMMA_F32_16X16X4_F32 (native fp32 matrix pipe, exact fp32
//     accumulation matching the reference).
//   * Selective scan: both directions concurrent in one launch, one lane per
//     (dir,b,d) channel, 16 register states, software-pipelined prefetch of
//     t+1 (global regs + 4-slot LDS B/C broadcast), 1 barrier per timestep.
//   * Workspace ~396 MiB via aliasing: delta reuses xz, yact aliases delta.
// ---------------------------------------------------------------------------

#define BATCH   8
#define SEQLEN  4096
#define LGL     12            // log2(SEQLEN)
#define D_MODEL 256
#define D_INNER 512
#define D_STATE 16
#define DT_RANK 16
#define MROWS   (BATCH * SEQLEN)   // 32768 GEMM rows

typedef __attribute__((ext_vector_type(2))) float v2f;
typedef __attribute__((ext_vector_type(8))) float v8f;

__device__ __forceinline__ v8f wmma_f32_16x16x4(v2f a, v2f b, v8f c) {
    // (neg_a, A, neg_b, B, c_mod, C, reuse_a, reuse_b)
    return __builtin_amdgcn_wmma_f32_16x16x4_f32(false, a, false, b,
                                                 (short)0, c, false, false);
}

__device__ __forceinline__ float silu_f(float x) {
    return x / (1.0f + __expf(-x));
}

// ---------------------------------------------------------------------------
// Generic fp32 WMMA GEMM:  C[m,n] (+)= scale * sum_k A[m,k] * W[n,k]
//   A: [MROWS, lda] activations (row m optionally time-reversed when revA)
//   W: [Ntot,  K]   weights, row-major; one 16-col tile staged in LDS
//   block = 256 threads (8 waves); each wave owns one 16x16 M-tile,
//   all 8 waves share the LDS weight tile -> 8x weight reuse, coalesced
//   activation loads (float2 per lane per k-step).
// ---------------------------------------------------------------------------
__global__ void __launch_bounds__(256)
gemm_wmma_kernel(const float* __restrict__ A, int lda,
                 const float* __restrict__ W,
                 float* __restrict__ C, int ldc,
                 int K, int kshift,
                 int revA, int revC, float scale, int accum)
{
    __shared__ float wt[D_INNER * 17]; // max K=512, pitch 17 (bank-conflict free)

    const int tid   = threadIdx.x;
    const int nbase = blockIdx.y << 4;

    // cooperative weight-tile load: 16 rows (n) x K cols, coalesced along k
    const int total = K << 4;
    const int kmask = K - 1;
    for (int idx = tid; idx < total; idx += 256) {
        const int j = idx >> kshift;
        const int k = idx & kmask;
        wt[k * 17 + j] = W[(size_t)(nbase + j) * K + k];
    }
    __syncthreads();

    const int wave = tid >> 5;
    const int lane = tid & 31;
    const int half = lane >> 4;
    const int lrow = lane & 15;
    const int kofs = half << 1;

    const int mbase = ((blockIdx.x << 3) + wave) << 4;
    const int arow  = mbase + lrow;
    int asrc = arow;
    if (revA) {
        const int b = arow >> LGL;
        const int l = arow & (SEQLEN - 1);
        asrc = (b << LGL) | (SEQLEN - 1 - l);
    }
    const float* __restrict__ Arow = A + (size_t)asrc * lda;

    v8f c = {};
    for (int k0 = 0; k0 < K; k0 += 4) {
        const v2f a = *(const v2f*)(Arow + k0 + kofs);   // A[m][k'], A[m][k'+1]
        v2f b;
        b.x = wt[(k0 + kofs    ) * 17 + lrow];           // B[k'  ][n]
        b.y = wt[(k0 + kofs + 1) * 17 + lrow];           // B[k'+1][n]
        c = wmma_f32_16x16x4(a, b, c);
    }

    #pragma unroll
    for (int i = 0; i < 8; ++i) {
        const int m = mbase + i + (half << 3);
        int orow = m;
        if (revC) {
            const int b = m >> LGL;
            const int l = m & (SEQLEN - 1);
            orow = (b << LGL) | (SEQLEN - 1 - l);
        }
        const size_t off = (size_t)orow * ldc + nbase + lrow;
        const float v = c[i] * scale;
        if (accum) C[off] += v; else C[off] = v;
    }
}

// ---------------------------------------------------------------------------
// Depthwise causal conv(4) + bias + SiLU on the x half of xz; SiLU on z half.
// ---------------------------------------------------------------------------
__global__ void __launch_bounds__(256)
conv_silu_kernel(const float* __restrict__ xz,
                 const float* __restrict__ cw,   // [512,4]
                 const float* __restrict__ cb,   // [512]
                 float* __restrict__ xs,         // [M,512]
                 float* __restrict__ zact)       // [M,512]
{
    const size_t idx = (size_t)blockIdx.x * 256 + threadIdx.x; // over M*512
    const int    d   = (int)(idx & (D_INNER - 1));
    const size_t row = idx >> 9;
    const int    l   = (int)(row & (SEQLEN - 1));

    const float w0 = cw[d * 4 + 0], w1 = cw[d * 4 + 1];
    const float w2 = cw[d * 4 + 2], w3 = cw[d * 4 + 3];
    const float* base = xz + row * (2 * D_INNER) + d;

    float acc = cb[d] + w3 * base[0];
    if (l >= 1) acc += w2 * base[-1 * 2 * D_INNER];
    if (l >= 2) acc += w1 * base[-2 * 2 * D_INNER];
    if (l >= 3) acc += w0 * base[-3 * 2 * D_INNER];

    xs[row * D_INNER + d] = silu_f(acc);

    const float z = xz[row * (2 * D_INNER) + D_INNER + d];
    zact[row * D_INNER + d] = silu_f(z);
}

// ---------------------------------------------------------------------------
// delta = softplus(dt @ dt_w.T + dt_b) via K=16 WMMA (4 x 16x16x4).
// One wave per 16x16 output tile.
// ---------------------------------------------------------------------------
__global__ void __launch_bounds__(32)
dt_softplus_kernel(const float* __restrict__ dbl,   // [M,48] (dt = cols 0..15)
                   const float* __restrict__ dtw,   // [512,16]
                   const float* __restrict__ dtb,   // [512]
                   float* __restrict__ delta)       // [M,512]
{
    const int lane = threadIdx.x;
    const int half = lane >> 4;
    const int lrow = lane & 15;
    const int kofs = half << 1;
    const int mbase = blockIdx.x << 4;
    const int nbase = blockIdx.y << 4;

    v8f c = {};
    #pragma unroll
    for (int k0 = 0; k0 < DT_RANK; k0 += 4) {
        const v2f a = *(const v2f*)(dbl + (size_t)(mbase + lrow) * 48 + k0 + kofs);
        const v2f b = *(const v2f*)(dtw + (size_t)(nbase + lrow) * DT_RANK + k0 + kofs);
        c = wmma_f32_16x16x4(a, b, c);
    }

    const float bias = dtb[nbase + lrow];
    #pragma unroll
    for (int i = 0; i < 8; ++i) {
        const int m = mbase + i + (half << 3);
        float v = c[i] + bias;
        v = (v > 20.0f) ? v : log1pf(__expf(v));       // softplus
        delta[(size_t)m * D_INNER + nbase + lrow] = v;
    }
}

// ---------------------------------------------------------------------------
// Selective scan, both directions in one launch (grid = 16 blocks: dir*8 + b).
// One lane per (dir,b,d): 16 register states, loop over L.
// Software pipelining: iteration t prefetches t+1's delta/xs/zact into
// registers and t+1's B/C vector into a 4-slot rotating LDS buffer, then
// computes t, then one barrier. 4 slots + 1 barrier/step is race-free:
// slot s is overwritten at iter s+3 (after sync(s+2)), read at iter s
// (before sync(s)).
// yact aliases delta: thread reads delta[row,d] (t and t+1) strictly before
// writing yact[row,d] in the same thread -> safe.
// Fuses the D skip-connection and the silu(z) gate.
// ---------------------------------------------------------------------------
__global__ void __launch_bounds__(512)
scan_dual_kernel(const float* __restrict__ xs,     // [2,B,L,512]
                 const float* __restrict__ delta,  // [2,B,L,512]
                 const float* __restrict__ dbl,    // [2,B,L,48] (B@16, C@32)
                 const float* __restrict__ zact,   // [2,B,L,512]
                 const float* __restrict__ A_log_f,
                 const float* __restrict__ A_log_b,
                 const float* __restrict__ D_f,
                 const float* __restrict__ D_b,
                 float* __restrict__ yact)         // [2,B,L,512] (== delta)
{
    __shared__ float bc[4][32];

    const int dir = blockIdx.x >> 3;
    const int b   = blockIdx.x & 7;
    const int d   = threadIdx.x;      // 0..511

    const size_t dofs = (size_t)dir * MROWS * D_INNER;
    const float* __restrict__ Xs = xs    + dofs;
    const float* __restrict__ Dl = delta + dofs;
    const float* __restrict__ Za = zact  + dofs;
    float*       __restrict__ Ya = yact  + dofs;
    const float* __restrict__ Db = dbl + (size_t)dir * MROWS * 48;
    const float* __restrict__ Alog = dir ? A_log_b : A_log_f;
    const float* __restrict__ Dp   = dir ? D_b     : D_f;

    float Arow[D_STATE];
    #pragma unroll
    for (int n = 0; n < D_STATE; ++n)
        Arow[n] = -__expf(Alog[d * D_STATE + n]);
    const float Dv = Dp[d];

    float h[D_STATE];
    #pragma unroll
    for (int n = 0; n < D_STATE; ++n) h[n] = 0.0f;

    const size_t base = (size_t)b << LGL;

    // prologue: stage t = 0
    if (threadIdx.x < 32)
        bc[0][threadIdx.x] = Db[base * 48 + DT_RANK + threadIdx.x];
    float dt = Dl[base * D_INNER + d];
    float u  = Xs[base * D_INNER + d];
    float za = Za[base * D_INNER + d];
    __syncthreads();

    for (int t = 0; t < SEQLEN; ++t) {
        const size_t row  = base + t;
        const int    tn   = (t + 1 < SEQLEN) ? (t + 1) : t;
        const size_t rown = base + tn;

        // --- prefetch t+1 (overlaps with the exp chain below) ---
        const float dt_n = Dl[rown * D_INNER + d];
        const float u_n  = Xs[rown * D_INNER + d];
        const float za_n = Za[rown * D_INNER + d];
        if (threadIdx.x < 32)
            bc[(t + 1) & 3][threadIdx.x] = Db[rown * 48 + DT_RANK + threadIdx.x];

        // --- compute t ---
        const float* Bc = bc[t & 3];
        const float* Cc = Bc + D_STATE;
        const float  du = dt * u;
        float acc = 0.0f;
        #pragma unroll
        for (int n = 0; n < D_STATE; ++n) {
            h[n] = __expf(dt * Arow[n]) * h[n] + du * Bc[n];
            acc += h[n] * Cc[n];
        }
        Ya[row * D_INNER + d] = (acc + u * Dv) * za;

        __syncthreads();
        dt = dt_n; u = u_n; za = za_n;
    }
}

// ---------------------------------------------------------------------------
// Launcher
// ---------------------------------------------------------------------------
extern "C" void kernel_launch(void* const* d_in, const int* in_sizes, int n_in,
                              void* d_out, int out_size, void* d_ws, size_t ws_size,
                              hipStream_t stream)
{
    (void)in_sizes; (void)n_in; (void)out_size; (void)ws_size;

    const float* u = (const float*)d_in[0];
    float* out = (float*)d_out;

    // Workspace layout (bytes), ~396 MiB total:
    //   [0, 128M)    xz scratch (per-dir, sequential) -> later delta[2] -> yact[2]
    //   [128M, 256M) xs[2]
    //   [256M, 384M) zact[2]
    //   [384M, 396M) dbl[2]
    char* ws = (char*)d_ws;
    const size_t HALF_B = (size_t)MROWS * D_INNER * 4;       //  64 MiB
    const size_t XZ_B   = 2 * HALF_B;                        // 128 MiB
    const size_t DBL_B  = (size_t)MROWS * 48 * 4;            //   6 MiB

    float* xz    = (float*)(ws);
    float* delta = (float*)(ws);                 // [2] dirs, aliases xz
    float* yact  = delta;                        // aliases delta (see scan)
    float* xs    = (float*)(ws + XZ_B);          // [2] dirs
    float* zact  = (float*)(ws + XZ_B + 2 * HALF_B);
    float* dbl   = (float*)(ws + XZ_B + 4 * HALF_B);

    const float* in_w[2], *conv_w[2], *conv_b[2], *xproj_w[2], *dt_w[2],
               * dt_b[2], *A_log[2], *Dp[2], *out_w[2];
    for (int dir = 0; dir < 2; ++dir) {
        const int o = dir ? 10 : 1;
        in_w[dir]    = (const float*)d_in[o + 0];
        conv_w[dir]  = (const float*)d_in[o + 1];
        conv_b[dir]  = (const float*)d_in[o + 2];
        xproj_w[dir] = (const float*)d_in[o + 3];
        dt_w[dir]    = (const float*)d_in[o + 4];
        dt_b[dir]    = (const float*)d_in[o + 5];
        A_log[dir]   = (const float*)d_in[o + 6];
        Dp[dir]      = (const float*)d_in[o + 7];
        out_w[dir]   = (const float*)d_in[o + 8];
    }

    // Phase A: in-projection + conv/silu per direction (xz region reused)
    for (int dir = 0; dir < 2; ++dir) {
        float* xs_d   = xs   + (size_t)dir * MROWS * D_INNER;
        float* zact_d = zact + (size_t)dir * MROWS * D_INNER;

        gemm_wmma_kernel<<<dim3(MROWS / 128, (2 * D_INNER) / 16), 256, 0, stream>>>(
            u, D_MODEL, in_w[dir], xz, 2 * D_INNER,
            D_MODEL, 8, /*revA=*/dir, /*revC=*/0, 1.0f, /*accum=*/0);

        conv_silu_kernel<<<(MROWS * D_INNER) / 256, 256, 0, stream>>>(
            xz, conv_w[dir], conv_b[dir], xs_d, zact_d);
    }

    // Phase B: x-projection + delta per direction (delta overwrites xz, now dead)
    for (int dir = 0; dir < 2; ++dir) {
        float* xs_d    = xs    + (size_t)dir * MROWS * D_INNER;
        float* dbl_d   = dbl   + (size_t)dir * MROWS * 48;
        float* delta_d = delta + (size_t)dir * MROWS * D_INNER;

        gemm_wmma_kernel<<<dim3(MROWS / 128, 3), 256, 0, stream>>>(
            xs_d, D_INNER, xproj_w[dir], dbl_d, DT_RANK + 2 * D_STATE,
            D_INNER, 9, 0, 0, 1.0f, 0);

        dt_softplus_kernel<<<dim3(MROWS / 16, D_INNER / 16), 32, 0, stream>>>(
            dbl_d, dt_w[dir], dt_b[dir], delta_d);
    }

    // Phase C: both directions' scans concurrently (the latency-bound stage)
    scan_dual_kernel<<<2 * BATCH, D_INNER, 0, stream>>>(
        xs, delta, dbl, zact,
        A_log[0], A_log[1], Dp[0], Dp[1], yact);

    // Phase D: out-projection; fwd stores 0.5*val, bwd adds 0.5*val reversed
    for (int dir = 0; dir < 2; ++dir) {
        float* yact_d = yact + (size_t)dir * MROWS * D_INNER;
        gemm_wmma_kernel<<<dim3(MROWS / 128, D_MODEL / 16), 256, 0, stream>>>(
            yact_d, D_INNER, out_w[dir], out, D_MODEL,
            D_INNER, 9, /*revA=*/0, /*revC=*/dir, 0.5f, /*accum=*/dir);
    }
}